// Codec_27273042330299
// MI455X (gfx1250) — compile-verified
//
#include <hip/hip_runtime.h>
#include <cstdint>
#include <cstddef>

// ---------------- problem geometry ----------------
#define WIDTH       1024
#define HEIGHT      1024
#define NIMG        8
#define STRIPS      64
#define ROWSPW      (HEIGHT / STRIPS)          // 16 rows per workgroup
#define HIST_U32    21504                      // per-channel-triple counters for 19 stages
#define REPL        2                          // LDS histogram replication (lane parity)
#define HIST_BYTES  (HIST_U32 * 4 * REPL)      // 172032
#define ROWBUF_OFF  ((uint32_t)HIST_BYTES)
#define ROWSTRIDE_B 12288u                     // 3 ch * 1024 * 4B per row slot
#define LDS_BYTES   (HIST_BYTES + 3 * 12288)   // 208896  (< 320KB WGP LDS)
#define GHIST_U32   172032                     // 19 stages * 24 channels * nbins

typedef __attribute__((ext_vector_type(2))) float v2f;
typedef __attribute__((ext_vector_type(4))) float v4f;
typedef __attribute__((ext_vector_type(8))) float v8f;

// per-stage base offsets: LDS layout [stage][c(3)][bin], global [stage][ch(24)][bin]
__constant__ int c_LB[19] = {0,1536,2304,3840,4608,6144,6912,8448,9216,10752,
                             11520,13056,13824,15360,16128,17664,18432,19968,20736};
__constant__ int c_GB[19] = {0,12288,18432,30720,36864,49152,55296,67584,73728,86016,
                             92160,104448,110592,122880,129024,141312,147456,159744,165888};

__device__ __forceinline__ int nbins_of_rt(int k) {
  return (k < 18 && ((k & 1) == 0)) ? 512 : 256;
}

// ---------------- CDNA5 async global->LDS row loader ----------------
__device__ __forceinline__ void issue_row(const float* __restrict__ x, int b, int y,
                                          int x0, uint32_t lds0) {
  const uint32_t slot = (uint32_t)(y % 3);
#pragma unroll
  for (int c = 0; c < 3; ++c) {
    const float* g = x + ((((size_t)b * 3 + c) * HEIGHT + y) * WIDTH + x0);
    uint32_t lo = lds0 + ROWBUF_OFF + slot * ROWSTRIDE_B + (uint32_t)c * 4096u
                + (uint32_t)x0 * 4u;
    // gfx1250 async path: per-lane 16B global -> LDS, tracked by ASYNCcnt
    asm volatile("global_load_async_to_lds_b128 %0, %1, off"
                 :: "v"(lo), "v"(g) : "memory");
  }
}

__device__ __forceinline__ void wait_async() {
  asm volatile("s_wait_asynccnt 0x0" ::: "memory");
}

// ---------------- exact fmod(v, 2.0f) for |v| << 2^24 (3 VALU ops) ----------------
// fmodf(v,2) = v - 2*trunc(v/2).  v*0.5 is exact, 2*trunc lies on v's ulp grid,
// so the single-rounding FMA returns the difference exactly (C trunc semantics, +-0 kept).
__device__ __forceinline__ float fast_fmod2(float v) {
  return __builtin_fmaf(-2.0f, truncf(v * 0.5f), v);
}

// ---------------- color transforms (all linear: 0 -> 0) ----------------
__device__ __forceinline__ void t_subg(float* s, float* t) {
  float r2 = s[0] - s[1], b2 = s[2] - s[1];
  t[0] = s[1]; t[1] = b2; t[2] = r2;
  s[0] = r2; s[2] = b2;                       // s[1] unchanged
}
__device__ __forceinline__ void t_jpeg(float* s, float* t) {
  float r2 = s[0] - s[1], b2 = s[2] - s[1], g2 = s[1] + (r2 + b2) * 0.25f;
  t[0] = g2; t[1] = b2; t[2] = r2;
  s[0] = r2; s[1] = g2; s[2] = b2;
}
__device__ __forceinline__ void t_ycocg(float* s, float* t) {
  float r2 = s[0] - s[2], b1 = s[2] + r2 * 0.5f, g2 = s[1] - b1, b2 = b1 + g2 * 0.5f;
  t[0] = b2; t[1] = g2; t[2] = r2;
  s[0] = r2; s[1] = g2; s[2] = b2;
}
template <int V>
__device__ __forceinline__ void t_ycbcr(float* s, float* t) {
  float r2 = s[0] - s[1], g1 = s[1] + r2 * 0.5f, b2 = s[2] - g1, u;
  if      (V == 1) u = b2 * 0.5f;
  else if (V == 2) u = (2.0f * b2 - r2) * 0.125f;
  else if (V == 3) u = (2.0f * b2 + r2) * 0.125f;
  else if (V == 4) u = b2 * (1.0f / 3.0f);
  else if (V == 5) u = b2 * 0.375f;
  else             u = b2 * 0.4375f;
  float g2 = g1 + u;
  t[0] = g2; t[1] = b2; t[2] = r2;
  s[0] = r2; s[1] = g2; s[2] = b2;
}

// one uniform scalar branch per 10-position batch (fi is wave-uniform)
__device__ __forceinline__ void chain_step(int fi, float (*sC)[3], float (*sN)[3],
                                           float (*tC)[3], float (*tN)[3]) {
#define ROWCASE(FI, FN)                                                      \
  case FI: {                                                                 \
    _Pragma("unroll")                                                        \
    for (int j = 0; j < 5; ++j) { FN(sC[j], tC[j]); FN(sN[j], tN[j]); }      \
  } break;
  switch (fi) {
    ROWCASE(0, t_subg)
    ROWCASE(1, t_jpeg)
    ROWCASE(2, t_ycocg)
    ROWCASE(3, t_ycbcr<1>)
    ROWCASE(4, t_ycbcr<2>)
    ROWCASE(5, t_ycbcr<3>)
    ROWCASE(6, t_ycbcr<4>)
    ROWCASE(7, t_ycbcr<5>)
    default: {
#pragma unroll
      for (int j = 0; j < 5; ++j) { t_ycbcr<6>(sC[j], tC[j]); t_ycbcr<6>(sN[j], tN[j]); }
    } break;
  }
#undef ROWCASE
}

__device__ __forceinline__ float wrap1(float v) { return fast_fmod2(v + 1.0f) - 1.0f; }

__device__ __forceinline__ float residf(float t, float N, float Wv, float NWv) {
  float mn = fminf(N, Wv), mx = fmaxf(N, Wv);
  float pred = fminf(fmaxf(N + Wv - NWv, mn), mx);
  return fast_fmod2(t - pred + 1.0f) - 1.0f;      // CALIC residual, C fmod semantics
}

__device__ __forceinline__ void push(unsigned* h, int off, int nb, float v, float lim) {
  if (v >= -lim && v <= lim) {                    // histc drops out-of-range
    int i = (int)floorf((v + lim) * 128.0f);      // scale == 128 for both bin configs
    i = i < 0 ? 0 : (i > nb - 1 ? nb - 1 : i);
    atomicAdd(&h[off + i], 1u);                   // LDS ds_add_u32
  }
}

__device__ __forceinline__ void hist_quad(unsigned* h, int base, int nb, float lim,
                                          const float (*tC)[3], const float (*tN)[3]) {
#pragma unroll
  for (int i = 1; i < 5; ++i) {                   // 4 center pixels of this thread
#pragma unroll
    for (int c = 0; c < 3; ++c) {
      float v = residf(tC[i][c], tN[i][c], tC[i - 1][c], tN[i - 1][c]);
      push(h, base + c * nb, nb, v, lim);
    }
  }
}

// ---------------- kernel 1: zero global histograms (every call) ----------------
__global__ void zero_u32(unsigned* __restrict__ g, int n) {
  int i = blockIdx.x * blockDim.x + threadIdx.x;
  if (i < n) g[i] = 0u;
}

// ---------------- kernel 2: fused transform chain + residual + histogram ----------------
__global__ void __launch_bounds__(256) rct_hist_kernel(const float* __restrict__ x,
                                                       unsigned* __restrict__ ghist) {
  extern __shared__ unsigned char smem[];
  unsigned* histL = (unsigned*)smem;
  const float* rowbuf = (const float*)(smem + ROWBUF_OFF);
  const int tid = threadIdx.x;
  const int b = blockIdx.x / STRIPS;
  const int strip = blockIdx.x % STRIPS;
  const int y0 = strip * ROWSPW;
  const int x0 = tid * 4;                        // 256 threads * 4 px = 1024-wide row
  const uint32_t lds0 = __builtin_amdgcn_groupstaticsize();
  unsigned* hmy = histL + (tid & (REPL - 1)) * HIST_U32;   // lane-parity replica

  for (int i = tid; i < HIST_U32 * REPL; i += 256) histL[i] = 0u;

  if (y0 > 0) issue_row(x, b, y0 - 1, x0, lds0);           // halo row from prev strip
  issue_row(x, b, y0, x0, lds0);
  wait_async();
  __syncthreads();

#pragma unroll 1
  for (int y = y0; y < y0 + ROWSPW; ++y) {
    if (y + 1 < y0 + ROWSPW) issue_row(x, b, y + 1, x0, lds0);   // prefetch 3rd slot

    // load 2 rows x 5 columns (x0-1..x0+3) x 3 channels of chain state from LDS
    float sC[5][3], sN[5][3], tC[5][3], tN[5][3];
    const int slC = y % 3;
#pragma unroll
    for (int c = 0; c < 3; ++c) {
      const float* rc = rowbuf + slC * 3072 + c * 1024;
      v4f v = *(const v4f*)(rc + x0);            // 16B-aligned ds_load_b128
      sC[0][c] = (x0 == 0) ? 0.0f : rc[x0 - 1];
      sC[1][c] = v.x; sC[2][c] = v.y; sC[3][c] = v.z; sC[4][c] = v.w;
    }
    if (y == 0) {
#pragma unroll
      for (int j = 0; j < 5; ++j) { sN[j][0] = 0.0f; sN[j][1] = 0.0f; sN[j][2] = 0.0f; }
    } else {
      const int slN = (y + 2) % 3;
#pragma unroll
      for (int c = 0; c < 3; ++c) {
        const float* rn = rowbuf + slN * 3072 + c * 1024;
        v4f v = *(const v4f*)(rn + x0);
        sN[0][c] = (x0 == 0) ? 0.0f : rn[x0 - 1];
        sN[1][c] = v.x; sN[2][c] = v.y; sN[3][c] = v.z; sN[4][c] = v.w;
      }
    }

    // 9 transforms x {raw pass (512 bins, +-2), wrapped pass (256 bins, +-1)}
    // rolled: keeps the hot loop ~I$-resident; fi branch is wave-uniform
#pragma unroll 1
    for (int fi = 0; fi < 9; ++fi) {
      chain_step(fi, sC, sN, tC, tN);
      hist_quad(hmy, c_LB[2 * fi], 512, 2.0f, tC, tN);
      chain_step(fi, sC, sN, tC, tN);
#pragma unroll
      for (int j = 0; j < 5; ++j)
#pragma unroll
        for (int c = 0; c < 3; ++c) { tC[j][c] = wrap1(tC[j][c]); tN[j][c] = wrap1(tN[j][c]); }
      hist_quad(hmy, c_LB[2 * fi + 1], 256, 1.0f, tC, tN);
    }
    // final stage: pred_cg on the mutated x itself
    hist_quad(hmy, c_LB[18], 256, 1.0f, sC, sN);

    wait_async();
    __syncthreads();
  }

  // merge replicated LDS histograms into global counters
#pragma unroll 1
  for (int k = 0; k < 19; ++k) {
    const int nb = nbins_of_rt(k);
    const int cnt = 3 * nb;
    const int lb = c_LB[k];
    const int gb = c_GB[k];
    for (int li = tid; li < cnt; li += 256) {
      unsigned v = histL[lb + li] + histL[HIST_U32 + lb + li];
      if (v) atomicAdd(&ghist[gb + (unsigned)(b * 3 + li / nb) * nb + (li % nb)], v);
    }
  }
}

// ---------------- kernel 3: entropy per stage (WMMA reduction) ----------------
__global__ void __launch_bounds__(256) rct_entropy_kernel(const unsigned* __restrict__ ghist,
                                                          float* __restrict__ out) {
  __shared__ float red[256];
  __shared__ float fold[64];
  const int k = blockIdx.x;
  const int nb = nbins_of_rt(k);
  const int base = c_GB[k];
  const int total = 24 * nb;

  float s = 0.0f;
  for (int i = threadIdx.x; i < total; i += 256) {
    unsigned cv = ghist[base + i];
    if (cv) {
      float p = (float)cv * (1.0f / 1048576.0f);   // res = 1024*1024
      s -= p * log2f(p);
    }
  }
  red[threadIdx.x] = s;
  __syncthreads();
  if (threadIdx.x < 64)
    fold[threadIdx.x] = red[threadIdx.x] + red[threadIdx.x + 64] +
                        red[threadIdx.x + 128] + red[threadIdx.x + 192];
  __syncthreads();

  if (threadIdx.x < 32) {               // wave 0 only: EXEC all-ones for WMMA
    const int l = threadIdx.x;
    v2f a;  a.x = fold[2 * l]; a.y = fold[2 * l + 1];   // 64 partials as 16x4 A
    v2f bo; bo.x = 1.0f; bo.y = 1.0f;                   // B = ones -> D[m,0]=row sums
    v8f c0 = {};
    v8f d = __builtin_amdgcn_wmma_f32_16x16x4_f32(false, a, false, bo,
                                                  (short)0, c0, false, false);
    float ds = d[0] + d[1] + d[2] + d[3] + d[4] + d[5] + d[6] + d[7];
    // column N=0 lives in lanes 0 (M=0..7) and 16 (M=8..15)
    float tot = __shfl(ds, 0, 32) + __shfl(ds, 16, 32);
    if (l == 0) out[k] = tot * 131072.0f;   // entropy/(8*24) * nelem, nelem=25165824
  }
}

// ---------------- launch ----------------
extern "C" void kernel_launch(void* const* d_in, const int* in_sizes, int n_in,
                              void* d_out, int out_size, void* d_ws, size_t ws_size,
                              hipStream_t stream) {
  const float* x = (const float*)d_in[0];
  unsigned* ghist = (unsigned*)d_ws;
  float* out = (float*)d_out;

  hipLaunchKernelGGL(zero_u32, dim3((GHIST_U32 + 255) / 256), dim3(256), 0, stream,
                     ghist, GHIST_U32);
  hipLaunchKernelGGL(rct_hist_kernel, dim3(NIMG * STRIPS), dim3(256), LDS_BYTES, stream,
                     x, ghist);
  hipLaunchKernelGGL(rct_entropy_kernel, dim3(19), dim3(256), 0, stream, ghist, out);

  (void)in_sizes; (void)n_in; (void)out_size; (void)ws_size;
}